// RateAdaptionDecoder_18940805775806
// MI455X (gfx1250) — compile-verified
//
#include <hip/hip_runtime.h>
#include <hip/hip_bf16.h>

typedef __attribute__((ext_vector_type(2))) float v2f;
typedef __attribute__((ext_vector_type(4))) float v4f;
typedef __attribute__((ext_vector_type(8))) float v8f;

#define RATE_NUM 8
#define MAX_RATE 256
#define C_DIM    256
#define B_DIM    4
#define L_DIM    1024                 // 32*32
#define NTOK     (B_DIM * L_DIM)      // 4096
#define MAXPERM  (NTOK + RATE_NUM*16) // 4224 (worst-case padded permutation)
#define MAXTILES (MAXPERM / 16)       // 264

// ws layout:
//   int   meta[0]              : numTokenTiles
//   int   meta[16..16+264)     : tileGroup[tile] -> rate index g
//   int   meta[512..512+4224)  : perm[] token ids sorted by group, pad = -1
//   float xT[4096][256]        : byte offset 32768   (token-major x)
//   float wT[8][256][256]      : byte offset 4227072 (column-major weight)
#define META_TILEGRP 16
#define META_PERM    512
#define XT_BYTE_OFF  32768
#define WT_BYTE_OFF  (XT_BYTE_OFF + NTOK * MAX_RATE * 4)   // 4227072

// ---------------------------------------------------------------------------
// Phase 1: bucket tokens by rate index; build padded permutation + tile map.
// Single block, 512 threads.
// ---------------------------------------------------------------------------
__global__ void group_tokens_kernel(const int* __restrict__ indexes,
                                    int* __restrict__ meta) {
    __shared__ int cnt[RATE_NUM];
    __shared__ int off[RATE_NUM];
    __shared__ int pofs[RATE_NUM + 1];
    const int tid = threadIdx.x;

    if (tid < RATE_NUM) { cnt[tid] = 0; off[tid] = 0; }
    __syncthreads();

    for (int t = tid; t < NTOK; t += 512)
        atomicAdd(&cnt[indexes[t] & 7], 1);
    __syncthreads();

    if (tid == 0) {
        int acc = 0;
        for (int g = 0; g < RATE_NUM; ++g) {
            pofs[g] = acc;
            acc += ((cnt[g] + 15) >> 4) << 4;   // pad each group to 16 tokens
        }
        pofs[RATE_NUM] = acc;
        meta[0] = acc >> 4;                      // number of 16-token tiles
    }
    __syncthreads();

    int* perm = meta + META_PERM;
    for (int i = tid; i < MAXPERM; i += 512) perm[i] = -1;
    __syncthreads();

    for (int t = tid; t < NTOK; t += 512) {
        int g = indexes[t] & 7;
        int p = atomicAdd(&off[g], 1);
        perm[pofs[g] + p] = t;
    }
    __syncthreads();

    int* tileGrp = meta + META_TILEGRP;
    const int nTiles = pofs[RATE_NUM] >> 4;
    for (int tile = tid; tile < nTiles; tile += 512) {
        int tok0 = tile << 4;
        int g = 0;
        for (int q = 0; q < RATE_NUM; ++q)
            if (tok0 >= pofs[q]) g = q;
        tileGrp[tile] = g;
    }
}

// ---------------------------------------------------------------------------
// Phase 2a: transpose x (B, R=256, L=1024) -> xT (token, R=256): A tiles
// become contiguous coalesced rows.
// ---------------------------------------------------------------------------
__global__ void transpose_x_kernel(const float* __restrict__ x,
                                   float* __restrict__ xT) {
    __shared__ float tile[32][33];
    const int b  = blockIdx.z;
    const int l0 = blockIdx.x * 32;
    const int r0 = blockIdx.y * 32;
    const int tx = threadIdx.x;            // 0..31
    const int ty = threadIdx.y;            // 0..7

    #pragma unroll
    for (int j = ty; j < 32; j += 8)
        tile[j][tx] = x[((b * MAX_RATE + r0 + j) << 10) + l0 + tx];
    __syncthreads();
    #pragma unroll
    for (int j = ty; j < 32; j += 8)
        xT[((size_t)(b * L_DIM + l0 + j) << 8) + r0 + tx] = tile[tx][j];
}

// ---------------------------------------------------------------------------
// Phase 2b: transpose weight (8, K=256, C=256) -> wT (8, C=256, K=256) so the
// GEMM B operand (per-lane K pair of a fixed column) is one global_load_b64
// instead of two strided global_load_b32.
// ---------------------------------------------------------------------------
__global__ void transpose_w_kernel(const float* __restrict__ w,
                                   float* __restrict__ wT) {
    __shared__ float tile[32][33];
    const int g  = blockIdx.z;
    const int c0 = blockIdx.x * 32;
    const int k0 = blockIdx.y * 32;
    const int tx = threadIdx.x;
    const int ty = threadIdx.y;

    #pragma unroll
    for (int j = ty; j < 32; j += 8)
        tile[j][tx] = w[(g << 16) + ((k0 + j) << 8) + c0 + tx];
    __syncthreads();
    #pragma unroll
    for (int j = ty; j < 32; j += 8)
        wT[(g << 16) + ((c0 + j) << 8) + k0 + tx] = tile[tx][j];
}

// ---------------------------------------------------------------------------
// Phase 3: grouped GEMM. One block = one 16-token tile; 16 waves cover the
// 16 channel tiles of N=256. A (16x256 f32, 16KB) staged once in LDS and
// shared by all waves; 64 x V_WMMA_F32_16X16X4_F32 per wave over K=256.
// Per WMMA step: 0.5 ds_load_2addr_b64 (A, 2 steps/op) + 1 global_load_b64 (B).
// ---------------------------------------------------------------------------
__global__ void __launch_bounds__(512)
rate_gemm_wmma_kernel(const float* __restrict__ xT,
                      const float* __restrict__ wT,
                      const float* __restrict__ bias,
                      const int* __restrict__ meta,
                      float* __restrict__ out) {
    __shared__ float As[16][258];   // +2 pad: conflict-light ds_load_b64
    __shared__ int   toks[16];
    __shared__ int   obase[16];     // per-token output base, -1 = padding row

    const int tt = blockIdx.x;
    if (tt >= meta[0]) return;                       // wave-uniform exit

    const int g     = meta[META_TILEGRP + tt];
    const int* perm = meta + META_PERM + (tt << 4);
    const int tid   = threadIdx.x;

    if (tid < 16) {
        int tok    = perm[tid];
        toks[tid]  = tok;
        // out(B,C,H,W): base = b*C*L + hw ; column adds col*1024
        obase[tid] = (tok < 0) ? -1 : (((tok >> 10) << 18) | (tok & 1023));
    }
    __syncthreads();

    // Cooperative A staging: 512 threads * 8 floats = 16 tokens * 256 K.
    {
        const int m    = tid >> 5;                   // token row 0..15
        const int lane = tid & 31;
        int tok  = toks[m];
        int tokc = tok < 0 ? 0 : tok;                // padded rows load row 0
        const float* src = xT + ((size_t)tokc << 8) + lane * 8;
        v4f v0 = *(const v4f*)(src);
        v4f v1 = *(const v4f*)(src + 4);
        float* dst = &As[m][lane * 8];
        *(v4f*)(dst)     = v0;                       // ds_store_b128
        *(v4f*)(dst + 4) = v1;
    }
    __syncthreads();

    const int lane = tid & 31;
    const int half = lane >> 4;                      // lane 16-31 => K+2 side
    const int ml   = lane & 15;                      // M (A) / N (B,C)
    const int col  = ((tid >> 5) << 4) + ml;         // channel 0..255

    // Column-major weight slice for this lane's column.
    const float* __restrict__ wgc = wT + ((size_t)g << 16) + ((size_t)col << 8);

    v8f acc = {};
    #pragma unroll 8
    for (int k0 = 0; k0 < MAX_RATE; k0 += 4) {
        const int ka = k0 + (half << 1);
        // A: VGPR0 = K=2*half, VGPR1 = K=2*half+1 (ds_load_b64, 8B aligned)
        v2f a = *(const v2f*)&As[ml][ka];
        // B: contiguous K pair of column `col` (global_load_b64, 8B aligned)
        v2f b = *(const v2f*)&wgc[ka];
        acc = __builtin_amdgcn_wmma_f32_16x16x4_f32(
            /*neg_a=*/false, a, /*neg_b=*/false, b,
            /*c_mod=*/(short)0, acc, /*reuse_a=*/false, /*reuse_b=*/false);
    }

    // C/D layout: lane holds column col; rows M = 8*half + r in acc[r].
    const float bv   = bias[(g << 8) + col];
    const int   colO = col << 10;                    // col * H*W
    #pragma unroll
    for (int r = 0; r < 8; ++r) {
        const int ob = obase[(half << 3) + r];
        if (ob >= 0)
            out[(size_t)(ob + colO)] = acc[r] + bv;
    }
}

// ---------------------------------------------------------------------------
extern "C" void kernel_launch(void* const* d_in, const int* in_sizes, int n_in,
                              void* d_out, int out_size, void* d_ws, size_t ws_size,
                              hipStream_t stream) {
    const float* x       = (const float*)d_in[0];
    const int*   indexes = (const int*)d_in[1];
    const float* weight  = (const float*)d_in[2];
    const float* bias    = (const float*)d_in[3];
    float* out = (float*)d_out;

    int*   meta = (int*)d_ws;
    float* xT   = (float*)((char*)d_ws + XT_BYTE_OFF);
    float* wT   = (float*)((char*)d_ws + WT_BYTE_OFF);

    group_tokens_kernel<<<1, 512, 0, stream>>>(indexes, meta);

    dim3 xgrid(L_DIM / 32, MAX_RATE / 32, B_DIM);    // (32, 8, 4)
    transpose_x_kernel<<<xgrid, dim3(32, 8), 0, stream>>>(x, xT);

    dim3 wgrid(C_DIM / 32, MAX_RATE / 32, RATE_NUM); // (8, 8, 8)
    transpose_w_kernel<<<wgrid, dim3(32, 8), 0, stream>>>(weight, wT);

    rate_gemm_wmma_kernel<<<MAXTILES, 512, 0, stream>>>(xT, wT, bias, meta, out);
}